// DownPooling_70927089926352
// MI455X (gfx1250) — compile-verified
//
#include <hip/hip_runtime.h>

// Cumulative max along W of [B=8, H=512, W=512, C=64] f32, layout [B,H,W,C].
// Memory-bound: 1 GiB traffic -> ~46 us floor at 23.3 TB/s.
// Strategy: one wave32 handles TWO (b,h) rows; lanes 0..15 -> row r
// (4 channels each, float4), lanes 16..31 -> row r+1. Each
// global_load_async_to_lds_b128 moves 512B = both rows' C-lines for one W
// position. 4-deep ASYNCcnt-tracked LDS ring per wave decouples the load
// stream from the sequential prefix-max dependency; no barriers since each
// wave consumes only LDS it produced.

constexpr int W_DIM = 512;
constexpr int C_DIM = 64;
constexpr int TP    = 8;            // W positions per tile
constexpr int NBUF  = 4;            // ring depth
constexpr int NT    = W_DIM / TP;   // 64 tiles
constexpr int WAVES = 8;            // waves per block (256 threads)
constexpr int LANES = 32;
constexpr int ROWS_PER_WAVE = 2;

__global__ __launch_bounds__(256)
void cummax_w_kernel(const float* __restrict__ in, float* __restrict__ out, int rows) {
    // per-wave ring: NBUF * TP positions * 512B = 16 KiB; x8 waves = 128 KiB
    __shared__ float4 smem[WAVES][NBUF * TP * LANES];

    const int wv   = threadIdx.x >> 5;
    const int lane = threadIdx.x & 31;
    const int half = lane >> 4;         // which of the wave's two rows
    const int sl   = lane & 15;         // sub-lane within the row's C-line
    const int row  = (blockIdx.x * WAVES + wv) * ROWS_PER_WAVE + half;
    if (row >= rows) return;

    const size_t rowbase = (size_t)row * (size_t)(W_DIM * C_DIM);
    const float* gbase = in  + rowbase + sl * 4;   // float4 slot in the C-line
    float*       obase = out + rowbase + sl * 4;

    // Low 32 bits of the generic pointer to a __shared__ object are the LDS
    // byte offset (flat aperture rule) -> usable as VDST for async-to-LDS.
    unsigned lds_base = (unsigned)(size_t)(&smem[wv][lane]);

    float4 run;
    run.x = run.y = run.z = run.w = -__builtin_huge_valf();

    // ---- prologue: fill NBUF tiles (TP async b128 ops per tile per lane) ----
#pragma unroll
    for (int j = 0; j < NBUF; ++j) {
        const float* ga = gbase + j * TP * C_DIM;
        unsigned     la = lds_base + (unsigned)(j * TP * 512);
#pragma unroll
        for (int ws = 0; ws < TP; ++ws) {
            asm volatile("global_load_async_to_lds_b128 %0, %1, off"
                         :: "v"(la), "v"(ga) : "memory");
            ga += C_DIM;   // next W position: +256 bytes per row in global
            la += 512;     // +512 bytes in LDS (two C-lines per position)
        }
    }

    // ---- steady-state pipeline ----
    for (int i = 0; i < NT; ++i) {
        if (i <= NT - NBUF) {
            // 4 tiles (32 async ops) in flight; allow 24 outstanding so the
            // oldest 8 (tile i) are guaranteed complete.
            asm volatile("s_wait_asynccnt 24" ::: "memory");
        } else {
            asm volatile("s_wait_asynccnt 0" ::: "memory");
        }

        // consume tile i from the ring
        const int bb = (i & (NBUF - 1)) * (TP * LANES);
        float* oa = obase + (size_t)i * TP * C_DIM;
#pragma unroll
        for (int ws = 0; ws < TP; ++ws) {
            float4 v = smem[wv][bb + ws * LANES + lane];
            run.x = fmaxf(run.x, v.x);
            run.y = fmaxf(run.y, v.y);
            run.z = fmaxf(run.z, v.z);
            run.w = fmaxf(run.w, v.w);
            *reinterpret_cast<float4*>(oa) = run;
            oa += C_DIM;
        }

        // refill: issue tile i+NBUF
        const int j = i + NBUF;
        if (j < NT) {
            const float* ga = gbase + j * TP * C_DIM;
            unsigned     la = lds_base + (unsigned)((j & (NBUF - 1)) * TP * 512);
#pragma unroll
            for (int ws = 0; ws < TP; ++ws) {
                asm volatile("global_load_async_to_lds_b128 %0, %1, off"
                             :: "v"(la), "v"(ga) : "memory");
                ga += C_DIM;
                la += 512;
            }
        }
    }
}

extern "C" void kernel_launch(void* const* d_in, const int* in_sizes, int n_in,
                              void* d_out, int out_size, void* d_ws, size_t ws_size,
                              hipStream_t stream) {
    const float* in  = (const float*)d_in[0];
    float*       out = (float*)d_out;
    const int total = in_sizes[0];             // B*H*W*C
    const int rows  = total / (W_DIM * C_DIM); // B*H = 4096
    const int rows_per_block = WAVES * ROWS_PER_WAVE;
    const int grid  = (rows + rows_per_block - 1) / rows_per_block;  // 256
    hipLaunchKernelGGL(cummax_w_kernel, dim3(grid), dim3(256), 0, stream, in, out, rows);
}